// InterAgg_20083267076374
// MI455X (gfx1250) — compile-verified
//
#include <hip/hip_runtime.h>
#include <math.h>

// Problem constants (from reference)
#define N_NODES 16384
#define KNB 32      // neighbors per relation
#define SSEL 16     // top-S selection
#define DF 190      // feature dim
#define DPAD 192    // padded feature dim (multiple of 4 for WMMA K)
#define EDIM 64     // embedding dim
#define NPAIRS (DPAD / 2)  // 96 K-pairs

typedef __attribute__((ext_vector_type(2))) float v2f;
typedef __attribute__((ext_vector_type(8))) float v8f;

__device__ __forceinline__ float wred(float v) {
#pragma unroll
  for (int off = 16; off > 0; off >>= 1) v += __shfl_xor(v, off, 32);
  return v;
}

// ---------------------------------------------------------------------------
// Kernel 1: IntraAgg. One wave32 per (node, relation).
// Writes padded rows into Xbuf: rows [0,N) = gathered center features,
// rows [(r+1)*N, (r+2)*N) = relation-r aggregated features. Cols 190,191 = 0.
// ---------------------------------------------------------------------------
__global__ __launch_bounds__(32) void intra_kernel(
    const float* __restrict__ feats, const int* __restrict__ nodes,
    const int* __restrict__ ng1, const int* __restrict__ ng2,
    const int* __restrict__ ng3, float* __restrict__ Xbuf) {
  const int n = blockIdx.x;
  const int r = blockIdx.y;
  const int lane = threadIdx.x;
  const int* ng = (r == 0) ? ng1 : (r == 1) ? ng2 : ng3;

  __shared__ float sim[KNB];
  __shared__ int nid[KNB];
  __shared__ int sel[KNB];

  nid[lane] = ng[(size_t)n * KNB + lane];

  const int cidx = nodes[n];
  const float* crow = feats + (size_t)cidx * DF;

  // center row: lane-over-dimension, 6 regs per lane
  float c[6];
  float csq = 0.f;
#pragma unroll
  for (int i = 0; i < 6; i++) {
    int d = lane + 32 * i;
    float v = (d < DF) ? crow[d] : 0.f;
    c[i] = v;
    csq += v * v;
  }
  csq = wred(csq);
  __syncthreads();

  // warm L2/L0 for the first neighbor row (gfx1250 global_prefetch_b8)
  __builtin_prefetch(feats + (size_t)nid[0] * DF + lane * 24, 0, 3);

  // cosine similarity per neighbor (coalesced lane-over-d loads)
  for (int k = 0; k < KNB; k++) {
    if (k + 1 < KNB) {  // prefetch next row while computing this one
      __builtin_prefetch(feats + (size_t)nid[k + 1] * DF + lane * 24, 0, 3);
    }
    const float* nrow = feats + (size_t)nid[k] * DF;
    float dot = 0.f, sq = 0.f;
#pragma unroll
    for (int i = 0; i < 6; i++) {
      int d = lane + 32 * i;
      float v = (d < DF) ? nrow[d] : 0.f;
      dot += c[i] * v;
      sq += v * v;
    }
    dot = wred(dot);
    sq = wred(sq);
    if (lane == 0) sim[k] = dot / sqrtf(csq * sq);
  }
  __syncthreads();

  // top-16 via rank counting; JAX tie-break: lower index wins
  float ms = sim[lane];
  int rank = 0;
#pragma unroll
  for (int j = 0; j < KNB; j++) {
    float s = sim[j];
    rank += (s > ms) || (s == ms && j < lane);
  }
  sel[lane] = (rank < SSEL) ? 1 : 0;
  __syncthreads();

  // masked mean of selected neighbor rows
  float acc[6] = {0.f, 0.f, 0.f, 0.f, 0.f, 0.f};
  for (int k = 0; k < KNB; k++) {
    if (sel[k]) {
      const float* nrow = feats + (size_t)nid[k] * DF;
#pragma unroll
      for (int i = 0; i < 6; i++) {
        int d = lane + 32 * i;
        if (d < DF) acc[i] += nrow[d];
      }
    }
  }

  float* xr = Xbuf + ((size_t)(r + 1) * N_NODES + n) * DPAD;
#pragma unroll
  for (int i = 0; i < 6; i++) {
    int d = lane + 32 * i;
    if (d < DPAD) xr[d] = (d < DF) ? fmaxf(acc[i] * (1.f / 16.f), 0.f) : 0.f;
  }
  if (r == 0) {
    float* cr = Xbuf + (size_t)n * DPAD;
#pragma unroll
    for (int i = 0; i < 6; i++) {
      int d = lane + 32 * i;
      if (d < DPAD) cr[d] = (d < DF) ? c[i] : 0.f;
    }
  }
}

// ---------------------------------------------------------------------------
// Kernel 2: H = X @ W via V_WMMA_F32_16X16X4_F32 (f32-exact).
// X: (4*N, 192) row-major. W (190,64) is staged in LDS repacked as K-pairs:
//   Wl2[p*64 + col] = (W[2p][col], W[2p+1][col])   (zero-padded past 190)
// so each B fragment (K rows {kb+2*hi, kb+2*hi+1} at one column) is a single
// aligned ds_load_b64 directly into the even VGPR pair WMMA consumes.
// A pair index == B pair index == kb/2 + hi, one global_load_b64 per K-step.
// Block = 128 threads (4 waves); block covers 64 M-rows; wave w owns rows
// [w*16, w*16+16) x all 64 cols (4 accumulator tiles). 48 K-steps of K=4.
// ---------------------------------------------------------------------------
__global__ __launch_bounds__(128) void gemm_wmma_kernel(
    const float* __restrict__ Xbuf, const float* __restrict__ weight,
    float* __restrict__ Hbuf) {
  __shared__ v2f Wl2[NPAIRS * EDIM];  // 48 KB
  const int tid = threadIdx.x;
  for (int idx = tid; idx < NPAIRS * EDIM; idx += 128) {
    const int p = idx >> 6;    // K-pair index 0..95
    const int col = idx & 63;  // output column
    const int k0 = 2 * p, k1 = 2 * p + 1;
    v2f w;
    w[0] = (k0 < DF) ? weight[k0 * EDIM + col] : 0.f;
    w[1] = (k1 < DF) ? weight[k1 * EDIM + col] : 0.f;
    Wl2[idx] = w;
  }
  __syncthreads();

  const int wave = tid >> 5;
  const int lane = tid & 31;
  const int m16 = lane & 15;
  const int hi = lane >> 4;  // 0 for lanes 0-15, 1 for lanes 16-31
  const size_t rowbase = (size_t)blockIdx.x * 64 + (size_t)wave * 16;

  v8f acc[4] = {};

  // A row for this lane, viewed as K-pairs (row stride 768 B, 8B-aligned)
  const v2f* arow2 = (const v2f*)(Xbuf + (rowbase + m16) * DPAD);

  for (int kb = 0; kb < DPAD; kb += 4) {
    const int ph = (kb >> 1) + hi;  // K-pair index for this lane half
    const v2f a = arow2[ph];        // A frag: (M=lane&15, K={kb+2hi, kb+2hi+1})
#pragma unroll
    for (int t = 0; t < 4; t++) {
      const v2f b = Wl2[ph * EDIM + t * 16 + m16];  // one ds_load_b64
      acc[t] = __builtin_amdgcn_wmma_f32_16x16x4_f32(
          false, a, false, b, (short)0, acc[t], false, false);
    }
  }

  // C/D layout: VGPR v -> M = v + 8*hi, N = lane&15
#pragma unroll
  for (int t = 0; t < 4; t++) {
#pragma unroll
    for (int v = 0; v < 8; v++) {
      const int m = v + 8 * hi;
      Hbuf[(rowbase + m) * EDIM + t * 16 + m16] = acc[t][v];
    }
  }
}

// ---------------------------------------------------------------------------
// Kernel 3: attention over 3 relations + combine. One wave32 per node,
// lane handles dims {lane, lane+32}. Block = 256 threads = 8 nodes.
// ---------------------------------------------------------------------------
__global__ __launch_bounds__(256) void attn_kernel(
    const float* __restrict__ Hbuf, const float* __restrict__ avec,
    float* __restrict__ out) {
  const int wave = threadIdx.x >> 5;
  const int lane = threadIdx.x & 31;
  const int node = blockIdx.x * 8 + wave;

  const float* ch = Hbuf + (size_t)node * EDIM;
  const float c0 = ch[lane];
  const float c1 = ch[lane + 32];
  const float dc = wred(c0 * avec[lane] + c1 * avec[lane + 32]);

  float e[3], nh0[3], nh1[3];
#pragma unroll
  for (int r = 0; r < 3; r++) {
    const float* nh = Hbuf + ((size_t)(r + 1) * N_NODES + node) * EDIM;
    nh0[r] = nh[lane];
    nh1[r] = nh[lane + 32];
    float dn = wred(nh0[r] * avec[EDIM + lane] + nh1[r] * avec[EDIM + lane + 32]);
    float x = dc + dn;
    e[r] = (x > 0.f) ? x : 0.2f * x;  // leaky_relu 0.2
  }
  const float mx = fmaxf(e[0], fmaxf(e[1], e[2]));
  const float w0 = expf(e[0] - mx);
  const float w1 = expf(e[1] - mx);
  const float w2 = expf(e[2] - mx);
  const float inv = 1.f / (w0 + w1 + w2);

  const float o0 = c0 + (w0 * nh0[0] + w1 * nh0[1] + w2 * nh0[2]) * inv;
  const float o1 = c1 + (w0 * nh1[0] + w1 * nh1[1] + w2 * nh1[2]) * inv;
  out[(size_t)node * EDIM + lane] = fmaxf(o0, 0.f);
  out[(size_t)node * EDIM + lane + 32] = fmaxf(o1, 0.f);
}

// ---------------------------------------------------------------------------
extern "C" void kernel_launch(void* const* d_in, const int* in_sizes, int n_in,
                              void* d_out, int out_size, void* d_ws,
                              size_t ws_size, hipStream_t stream) {
  const float* feats = (const float*)d_in[0];   // (100000, 190) f32
  const int* nodes = (const int*)d_in[1];       // (16384,)
  const int* ng1 = (const int*)d_in[2];         // (16384, 32)
  const int* ng2 = (const int*)d_in[3];
  const int* ng3 = (const int*)d_in[4];
  const float* weight = (const float*)d_in[5];  // (190, 64)
  const float* avec = (const float*)d_in[6];    // (128, 1)

  float* Xbuf = (float*)d_ws;                       // (4*N, 192) f32
  float* Hbuf = Xbuf + (size_t)4 * N_NODES * DPAD;  // (4*N, 64) f32
  float* out = (float*)d_out;                       // (N, 64) f32

  intra_kernel<<<dim3(N_NODES, 3), 32, 0, stream>>>(feats, nodes, ng1, ng2,
                                                    ng3, Xbuf);
  gemm_wmma_kernel<<<(4 * N_NODES) / 64, 128, 0, stream>>>(Xbuf, weight, Hbuf);
  attn_kernel<<<N_NODES / 8, 256, 0, stream>>>(Hbuf, avec, out);
}